// TaskAlignedAssigner_67645734912000
// MI455X (gfx1250) — compile-verified
//
#include <hip/hip_runtime.h>
#include <hip/hip_bf16.h>
#include <math.h>
#include <stdint.h>

#define BS 16
#define NA 8400
#define NC 80
#define NG 128
#define TOPK_K 13
#define ATILES 525   /* NA/16 exactly */
#define EPS_IN 1e-9f

typedef float v2f __attribute__((ext_vector_type(2)));
typedef float v8f __attribute__((ext_vector_type(8)));
typedef unsigned int u32x4 __attribute__((ext_vector_type(4)));
typedef int i32x8 __attribute__((ext_vector_type(8)));
typedef int i32x4 __attribute__((ext_vector_type(4)));

// CIoU clipped at 0, replicating the reference (eps = 1e-7)
__device__ __forceinline__ float ciou_pos(float b1x1, float b1y1, float b1x2, float b1y2,
                                          float b2x1, float b2y1, float b2x2, float b2y2) {
    const float eps = 1e-7f;
    float iw = fminf(b1x2, b2x2) - fmaxf(b1x1, b2x1); iw = fmaxf(iw, 0.f);
    float ih = fminf(b1y2, b2y2) - fmaxf(b1y1, b2y1); ih = fmaxf(ih, 0.f);
    float inter = iw * ih;
    float w1 = b1x2 - b1x1, h1 = b1y2 - b1y1 + eps;
    float w2 = b2x2 - b2x1, h2 = b2y2 - b2y1 + eps;
    float uni = w1 * h1 + w2 * h2 - inter + eps;
    float iou = inter / uni;
    float cw = fmaxf(b1x2, b2x2) - fminf(b1x1, b2x1);
    float ch = fmaxf(b1y2, b2y2) - fminf(b1y1, b2y1);
    float c2 = cw * cw + ch * ch + eps;
    float dx = b2x1 + b2x2 - b1x1 - b1x2;
    float dy = b2y1 + b2y2 - b1y1 - b1y2;
    float rho2 = (dx * dx + dy * dy) * 0.25f;
    float t = atanf(w2 / h2) - atanf(w1 / h1);
    float v = 0.40528473456935108f * t * t;  // 4/pi^2
    float alpha = v / (v - iou + (1.0f + eps));
    float c = iou - (rho2 / c2 + v * alpha);
    return fmaxf(c, 0.f);
}

// ---------------------------------------------------------------------------
// K1: one-hot GEMM via V_WMMA_F32_16X16X4_F32 (exact gather), fused with CIoU
// and in-gt mask -> align[b][a][g]  (layout: ((b*NA+a)*NG + g))
// Block = 256 threads = 8 waves; wave w handles gt-tile w (NG = 8*16).
// Grid = BS*ATILES; block covers 16 anchors.
// ---------------------------------------------------------------------------
__global__ __launch_bounds__(256) void k1_align(
    const float* __restrict__ pd_scores, const float* __restrict__ pd_bboxes,
    const float* __restrict__ anc_points, const int* __restrict__ gt_labels,
    const float* __restrict__ gt_bboxes, const float* __restrict__ gt_mask,
    float* __restrict__ alignBuf) {
    __shared__ float sA[16 * NC];       // 16 anchors x 80 classes
    __shared__ float sPB[16 * 4];       // 16 pred boxes
    __shared__ float sGtB[NG * 4];      // 128 gt boxes
    __shared__ int   sLab[NG];
    __shared__ float sMask[NG];

    const int bx = blockIdx.x;
    const int b  = bx / ATILES;
    const int a0 = (bx % ATILES) * 16;
    const int tid = threadIdx.x;

    for (int i = tid; i < 16 * NC; i += 256) {
        int m = i / NC, c = i % NC;
        sA[i] = pd_scores[((size_t)b * NA + (a0 + m)) * NC + c];
    }
    for (int i = tid; i < 16 * 4; i += 256)
        sPB[i] = pd_bboxes[((size_t)b * NA + a0) * 4 + i];
    for (int i = tid; i < NG * 4; i += 256)
        sGtB[i] = gt_bboxes[((size_t)b * NG) * 4 + i];
    for (int i = tid; i < NG; i += 256) {
        sLab[i]  = gt_labels[b * NG + i];
        sMask[i] = gt_mask[b * NG + i];
    }
    __syncthreads();

    const int wv    = tid >> 5;     // gt tile 0..7
    const int lane  = tid & 31;
    const int lhalf = lane >> 4;    // 0/1
    const int lm    = lane & 15;
    const int g_lane = wv * 16 + lm;
    const int lab    = sLab[g_lane];

    v8f acc = {};
    #pragma unroll
    for (int k = 0; k < NC; k += 4) {
        const int kk = k + 2 * lhalf;
        v2f af, bf;
        af.x = sA[lm * NC + kk];
        af.y = sA[lm * NC + kk + 1];
        bf.x = (lab == kk)     ? 1.0f : 0.0f;
        bf.y = (lab == kk + 1) ? 1.0f : 0.0f;
        acc = __builtin_amdgcn_wmma_f32_16x16x4_f32(
            /*neg_a=*/false, af, /*neg_b=*/false, bf,
            /*c_mod=*/(short)0, acc, /*reuse_a=*/false, /*reuse_b=*/false);
    }

    // Epilogue: acc[r] = pd_scores[b, a0+m, lab] for m = r + 8*lhalf
    const float gx1 = sGtB[g_lane * 4 + 0], gy1 = sGtB[g_lane * 4 + 1];
    const float gx2 = sGtB[g_lane * 4 + 2], gy2 = sGtB[g_lane * 4 + 3];
    const float gm  = sMask[g_lane];
    #pragma unroll
    for (int r = 0; r < 8; ++r) {
        const int m = r + 8 * lhalf;
        const int a = a0 + m;
        const float px1 = sPB[m * 4 + 0], py1 = sPB[m * 4 + 1];
        const float px2 = sPB[m * 4 + 2], py2 = sPB[m * 4 + 3];
        const float ax = anc_points[2 * a], ay = anc_points[2 * a + 1];
        float dmin = fminf(fminf(ax - gx1, ay - gy1), fminf(gx2 - ax, gy2 - ay));
        float maskpos = (dmin > EPS_IN) ? gm : 0.0f;
        float iou = ciou_pos(px1, py1, px2, py2, gx1, gy1, gx2, gy2);
        float i2 = iou * iou;
        float i6 = i2 * i2 * i2;
        float s = acc[r];                 // exact gathered score
        alignBuf[((size_t)b * NA + a) * NG + g_lane] = s * i6 * maskpos;
    }
}

// ---------------------------------------------------------------------------
// K2: top-13 over anchors per (b,g) via 13 block-argmax passes
// (key order: value desc, index asc == jax.lax.top_k tie-break)
// Grid = BS*NG blocks, 256 threads.
// ---------------------------------------------------------------------------
__global__ __launch_bounds__(256) void k2_topk(
    const float* __restrict__ alignBuf, const float* __restrict__ gt_mask,
    uint8_t* __restrict__ isin) {
    __shared__ float sv[256];
    __shared__ int   si[256];
    const int bg = blockIdx.x;
    const int b = bg / NG, g = bg % NG;
    const int tid = threadIdx.x;
    const float gm = gt_mask[b * NG + g];
    const float* arow = alignBuf + (size_t)b * NA * NG + g;  // stride NG

    float prevVal = __builtin_inff();
    int   prevIdx = -1;
    for (int k = 0; k < TOPK_K; ++k) {
        float best = -__builtin_inff();
        int bidx = NA;
        for (int a = tid; a < NA; a += 256) {
            float v = arow[(size_t)a * NG];
            bool excluded = (v > prevVal) || (v == prevVal && a <= prevIdx);
            if (!excluded && (v > best || (v == best && a < bidx))) { best = v; bidx = a; }
        }
        sv[tid] = best; si[tid] = bidx;
        __syncthreads();
        for (int off = 128; off > 0; off >>= 1) {
            if (tid < off) {
                float v2 = sv[tid + off]; int i2 = si[tid + off];
                if (v2 > sv[tid] || (v2 == sv[tid] && i2 < si[tid])) { sv[tid] = v2; si[tid] = i2; }
            }
            __syncthreads();
        }
        float wvv = sv[0]; int wii = si[0];
        __syncthreads();
        prevVal = wvv; prevIdx = wii;
        if (tid == 0 && wii < NA) {
            isin[((size_t)b * NA + wii) * NG + g] =
                (gm > 0.0f && wvv > 1e-9f) ? (uint8_t)1 : (uint8_t)0;
        }
    }
}

// ---------------------------------------------------------------------------
// K3: set global conflict flag if any anchor selected by >1 gt
// ---------------------------------------------------------------------------
__global__ __launch_bounds__(256) void k3_conflict(
    const uint8_t* __restrict__ isin, int* __restrict__ flag) {
    const int idx = blockIdx.x * 256 + threadIdx.x;
    if (idx >= BS * NA) return;
    const unsigned int* r = (const unsigned int*)(isin + (size_t)idx * NG);
    unsigned int acc = 0;
    #pragma unroll
    for (int i = 0; i < NG / 4; ++i) acc += r[i];   // per-byte sums <= 32, no carry
    unsigned int total = (acc & 0xFF) + ((acc >> 8) & 0xFF) + ((acc >> 16) & 0xFF) + (acc >> 24);
    if (total > 1) atomicOr(flag, 1);
}

// ---------------------------------------------------------------------------
// K4: per-anchor conflict resolution + targets + norm_align.
// GT boxes staged into LDS via Tensor Data Mover (TDM).
// Grid = (ceil(NA/256), BS)
// ---------------------------------------------------------------------------
__global__ __launch_bounds__(256) void k4_finalize(
    const float* __restrict__ pd_bboxes, const float* __restrict__ gt_bboxes,
    const int* __restrict__ gt_labels, const float* __restrict__ alignBuf,
    const uint8_t* __restrict__ isin, const int* __restrict__ flag,
    float* __restrict__ out) {
    __shared__ float sGtB[NG * 4];
    __shared__ int   sLab[NG];
    const int b = blockIdx.y;
    const int tid = threadIdx.x;

#if __has_builtin(__builtin_amdgcn_tensor_load_to_lds)
    if (tid == 0) {
        // Tensor DMA descriptor: 1-D tensor of 512 f32 (128 gt boxes) -> LDS
        uint64_t ga = (uint64_t)(uintptr_t)(gt_bboxes + (size_t)b * NG * 4);
        uint32_t ldsOff = (uint32_t)(uintptr_t)(void*)sGtB;  // low 32b of LDS aperture = LDS offset
        u32x4 g0;
        g0.x = 1u;                                   // count=1 valid descriptor
        g0.y = ldsOff;                               // lds_addr [63:32]
        g0.z = (uint32_t)(ga & 0xFFFFFFFFu);         // global_addr [95:64]
        g0.w = (uint32_t)((ga >> 32) & 0x1FFFFFFu) | (2u << 30);  // addr hi + type=2
        i32x8 g1;
        g1[0] = (int)(2u << 16);    // data_size=4B; no multicast/iterate/pad
        g1[1] = (int)(512u << 16);  // tensor_dim0 lo16 at [63:48]
        g1[2] = (int)(1u << 16);    // tensor_dim1 = 1 at [95:80]
        g1[3] = (int)(512u << 16);  // tile_dim0 = 512 at [127:112]
        g1[4] = 1;                  // tile_dim1 = 1
        g1[5] = 512;                // tensor_dim0_stride
        g1[6] = 0; g1[7] = 0;
        i32x4 z4 = {0, 0, 0, 0};
        i32x8 z8 = {0, 0, 0, 0, 0, 0, 0, 0};
        __builtin_amdgcn_tensor_load_to_lds(g0, g1, z4, z4, z8, 0);
    }
#else
    for (int i = tid; i < NG * 4; i += 256)
        sGtB[i] = gt_bboxes[((size_t)b * NG) * 4 + i];
#endif
    for (int i = tid; i < NG; i += 256) sLab[i] = gt_labels[b * NG + i];
#if __has_builtin(__builtin_amdgcn_s_wait_tensorcnt)
    __builtin_amdgcn_s_wait_tensorcnt(0);
#endif
    __syncthreads();

    const int a = blockIdx.x * 256 + tid;
    if (a >= NA) return;
    const size_t ba = (size_t)b * NA + a;
    const float px1 = pd_bboxes[ba * 4 + 0], py1 = pd_bboxes[ba * 4 + 1];
    const float px2 = pd_bboxes[ba * 4 + 2], py2 = pd_bboxes[ba * 4 + 3];
    const uint8_t* row = isin + ba * NG;
    const float* arow = alignBuf + ba * NG;
    const bool anyConf = (*flag) != 0;

    // pass 1: is_in sum + best_gt = argmax(align) (first occurrence)
    int sum = 0, bestG = 0;
    float bestAlign = -__builtin_inff();
    for (int g = 0; g < NG; ++g) {
        sum += row[g];
        float al = arow[g];
        if (al > bestAlign) { bestAlign = al; bestG = g; }
    }
    const bool myConf = sum > 1;

    // pass 2: resolved values, argmax, max(align*v), max(iou*v)
    int maxV = 0, tgi = 0;
    float maxAmf = 0.0f, maxIouW = 0.0f;
    for (int g = 0; g < NG; ++g) {
        int e = row[g];
        int v = anyConf ? (myConf ? 0 : e + (g == bestG ? 1 : 0)) : e;
        if (v > maxV) { maxV = v; tgi = g; }
        if (v > 0) {
            float vf = (float)v;
            float amf = arow[g] * vf;
            if (amf > maxAmf) maxAmf = amf;
            float iou = ciou_pos(px1, py1, px2, py2,
                                 sGtB[g * 4 + 0], sGtB[g * 4 + 1],
                                 sGtB[g * 4 + 2], sGtB[g * 4 + 3]);
            float iw = iou * vf;
            if (iw > maxIouW) maxIouW = iw;
        }
    }
    const bool fmask = anyConf ? (!myConf) : (sum > 0);
    const float norm = maxAmf * maxAmf / (maxIouW + 1e-9f);

    float* out_cls  = out;
    float* out_bbox = out + (size_t)BS * NA;
    float* out_mask = out + (size_t)BS * NA * 5 + (size_t)BS * NA * NC;
    float* out_norm = out_mask + (size_t)BS * NA;
    out_cls[ba] = (float)sLab[tgi];
    out_bbox[ba * 4 + 0] = sGtB[tgi * 4 + 0];
    out_bbox[ba * 4 + 1] = sGtB[tgi * 4 + 1];
    out_bbox[ba * 4 + 2] = sGtB[tgi * 4 + 2];
    out_bbox[ba * 4 + 3] = sGtB[tgi * 4 + 3];
    out_mask[ba] = fmask ? 1.0f : 0.0f;
    out_norm[ba] = norm;
}

// ---------------------------------------------------------------------------
// K5: target_scores = one_hot(target_cls, 80) * norm_align (coalesced)
// ---------------------------------------------------------------------------
__global__ __launch_bounds__(256) void k5_scores(float* __restrict__ out) {
    const size_t idx = (size_t)blockIdx.x * 256 + threadIdx.x;
    if (idx >= (size_t)BS * NA * NC) return;
    const size_t ba = idx / NC;
    const int c = (int)(idx % NC);
    const float* out_cls  = out;
    const float* out_norm = out + (size_t)BS * NA * 5 + (size_t)BS * NA * NC + (size_t)BS * NA;
    float* out_scores = out + (size_t)BS * NA * 5;
    int cls = (int)out_cls[ba];
    out_scores[idx] = (c == cls) ? out_norm[ba] : 0.0f;
}

extern "C" void kernel_launch(void* const* d_in, const int* in_sizes, int n_in,
                              void* d_out, int out_size, void* d_ws, size_t ws_size,
                              hipStream_t stream) {
    const float* pd_scores  = (const float*)d_in[0];
    const float* pd_bboxes  = (const float*)d_in[1];
    const float* anc_points = (const float*)d_in[2];
    const int*   gt_labels  = (const int*)d_in[3];
    const float* gt_bboxes  = (const float*)d_in[4];
    const float* gt_mask    = (const float*)d_in[5];
    float* out = (float*)d_out;

    const size_t alignElems = (size_t)BS * NA * NG;
    float*   alignBuf = (float*)d_ws;
    uint8_t* isin     = (uint8_t*)d_ws + alignElems * sizeof(float);
    int*     flag     = (int*)(isin + alignElems);

    (void)hipMemsetAsync(isin, 0, alignElems + sizeof(int), stream);

    k1_align<<<BS * ATILES, 256, 0, stream>>>(pd_scores, pd_bboxes, anc_points,
                                              gt_labels, gt_bboxes, gt_mask, alignBuf);
    k2_topk<<<BS * NG, 256, 0, stream>>>(alignBuf, gt_mask, isin);
    k3_conflict<<<(BS * NA + 255) / 256, 256, 0, stream>>>(isin, flag);
    k4_finalize<<<dim3((NA + 255) / 256, BS), 256, 0, stream>>>(
        pd_bboxes, gt_bboxes, gt_labels, alignBuf, isin, flag, out);
    k5_scores<<<(int)(((size_t)BS * NA * NC + 255) / 256), 256, 0, stream>>>(out);
}